// QueryDynamicAttention_29686813950617
// MI455X (gfx1250) — compile-verified
//
#include <hip/hip_runtime.h>
#include <hip/hip_bf16.h>
#include <math.h>

// ---------------------------------------------------------------------------
// Problem constants (from reference): B=16, N=1024, D=768, MU=768, DR=48
// ---------------------------------------------------------------------------
#define BSZ   16
#define NTOK  1024
#define DIM   768
#define MUD   768
#define DRED  48
#define NT_BIG (DRED * DIM)   // 36864

typedef float v2f __attribute__((ext_vector_type(2)));
typedef float v8f __attribute__((ext_vector_type(8)));

// ---------------------------------------------------------------------------
// Kernel 1: hyper-network GEMM via f32 WMMA.
//   C[16, Nt] = mu[16, 768] @ Wg[768, Nt] + bg[Nt]
// One wave computes one 16x16 output tile; 8 waves (256 threads) per block.
// Nt must be a multiple of 128. mu is staged in LDS (48 KB).
//
// V_WMMA_F32_16X16X4_F32 operand layouts (wave32):
//   A 16x4 f32 (2 VGPRs/lane): lanes 0-15 hold M=lane, K={k0,k0+1};
//                              lanes 16-31 hold M=lane-16, K={k0+2,k0+3}.
//   B 4x16 f32 (2 VGPRs/lane): lanes 0-15 N=lane K={k0,k0+1};
//                              lanes 16-31 N=lane-16 K={k0+2,k0+3}.
//   C/D 16x16 f32 (8 VGPRs):   VGPR v, lane group hi: M = v + 8*hi, N = lane%16.
// ---------------------------------------------------------------------------
__global__ __launch_bounds__(256) void hyper_gemm_wmma(
    const float* __restrict__ mu,   // [16, 768]
    const float* __restrict__ Wg,   // [768, Nt]
    const float* __restrict__ bg,   // [Nt]
    float* __restrict__ Cout,       // [16, Nt]
    int Nt)
{
    __shared__ float smu[BSZ * MUD];   // 48 KB

    const int t = threadIdx.x;
    for (int j = t; j < BSZ * MUD; j += 256) smu[j] = mu[j];
    __syncthreads();

    const int wave = t >> 5;
    const int lane = t & 31;
    const int hi   = lane >> 4;     // 0 or 1
    const int lm   = lane & 15;

    const int tile  = blockIdx.x * 8 + wave;
    const int n     = tile * 16 + lm;

    v8f acc = {0.f, 0.f, 0.f, 0.f, 0.f, 0.f, 0.f, 0.f};

    const float* arow = &smu[lm * MUD + 2 * hi];
    const float* bcol = Wg + (size_t)(2 * hi) * (size_t)Nt + (size_t)n;

    #pragma unroll 4
    for (int k0 = 0; k0 < MUD; k0 += 4) {
        // A fragment from LDS (8B-aligned: even index) -> ds_load_b64
        v2f a = *reinterpret_cast<const v2f*>(arow + k0);
        // B fragment: two rows of Wg, non-temporal (pure stream, 113 MB)
        const float* bp = bcol + (size_t)k0 * (size_t)Nt;
        v2f bf;
        bf.x = __builtin_nontemporal_load(bp);
        bf.y = __builtin_nontemporal_load(bp + Nt);
        acc = __builtin_amdgcn_wmma_f32_16x16x4_f32(
            /*neg_a=*/false, a, /*neg_b=*/false, bf,
            /*c_mod=*/(short)0, acc, /*reuse_a=*/false, /*reuse_b=*/false);
    }

    const float bgv = bg[n];
    #pragma unroll
    for (int v = 0; v < 8; ++v) {
        const int m = v + 8 * hi;
        Cout[(size_t)m * (size_t)Nt + (size_t)n] = acc[v] + bgv;
    }
}

// ---------------------------------------------------------------------------
// Kernel 2: avg + max pooling over the token dimension.
// One thread per (b, d) column; consecutive threads -> consecutive d (coalesced).
// Grid: (B*D)/256 = 48 blocks.
// ---------------------------------------------------------------------------
__global__ __launch_bounds__(256) void pool_kernel(
    const float* __restrict__ x,    // [B, N, D]
    float* __restrict__ avg,        // [B, D]
    float* __restrict__ mxp)        // [B, D]
{
    const int gid = blockIdx.x * 256 + threadIdx.x;   // 0 .. B*D-1
    const int b = gid / DIM;
    const int d = gid - b * DIM;
    const float* p = x + (size_t)b * NTOK * DIM + d;
    float s = 0.f;
    float m = -__builtin_inff();
    for (int nn = 0; nn < NTOK; ++nn) {
        const float v = p[(size_t)nn * DIM];
        s += v;
        m = fmaxf(m, v);
    }
    avg[gid] = s * (1.0f / (float)NTOK);
    mxp[gid] = m;
}

// ---------------------------------------------------------------------------
// Kernel 3: per-sample dynamic MLP + channel gate.
// One block per sample b. Computes:
//   h_p = relu(pool_p @ W1[b]^T + (mu@Bg1 + bb1))  for p in {avg, max}
//   att = (h_avg + h_max) @ W2[b]^T + 2*(mu@Bg2 + bb2)   (layer 2 is linear)
//   gate[b,:] = sigmoid(att);  bs[b] = mu@Bgs + bbs
// ---------------------------------------------------------------------------
__global__ __launch_bounds__(256) void mlp_gate_kernel(
    const float* __restrict__ mu,     // [B, MU]
    const float* __restrict__ W1,     // [B, DR*D]  (W1[b,o,i] = W1[b, o*D+i])
    const float* __restrict__ W2,     // [B, D*DR]  (W2[b,o,h] = W2[b, o*DR+h])
    const float* __restrict__ Bg1,    // [MU, DR]
    const float* __restrict__ bb1,    // [DR]
    const float* __restrict__ Bg2,    // [MU, D]
    const float* __restrict__ bb2,    // [D]
    const float* __restrict__ Bgs,    // [MU, 1]
    const float* __restrict__ bbs,    // [1]
    const float* __restrict__ avg,    // [B, D]
    const float* __restrict__ mxp,    // [B, D]
    float* __restrict__ gate,         // [B, D]
    float* __restrict__ bs_out)       // [B]
{
    __shared__ float h2[2][DRED];

    const int b = blockIdx.x;
    const int t = threadIdx.x;
    const float* murow = mu + b * MUD;

    if (t == 0) {
        float acc = 0.f;
        for (int m = 0; m < MUD; ++m) acc += murow[m] * Bgs[m];
        bs_out[b] = acc + bbs[0];
    }

    if (t < 2 * DRED) {
        const int pool = t / DRED;
        const int o    = t - pool * DRED;
        const float* pv   = (pool == 0 ? avg : mxp) + b * DIM;
        const float* wrow = W1 + (size_t)b * NT_BIG + (size_t)o * DIM;
        float acc = 0.f;
        for (int i = 0; i < DIM; ++i) acc += pv[i] * wrow[i];
        float b1 = 0.f;
        for (int m = 0; m < MUD; ++m) b1 += murow[m] * Bg1[m * DRED + o];
        h2[pool][o] = fmaxf(acc + b1 + bb1[o], 0.f);
    }
    __syncthreads();

    for (int o2 = t; o2 < DIM; o2 += 256) {
        float b2 = 0.f;
        for (int m = 0; m < MUD; ++m) b2 += murow[m] * Bg2[m * DIM + o2];
        b2 += bb2[o2];
        const float* w2row = W2 + (size_t)b * NT_BIG + (size_t)o2 * DRED;
        float att = 0.f;
        #pragma unroll
        for (int i = 0; i < DRED; ++i) att += (h2[0][i] + h2[1][i]) * w2row[i];
        att += 2.f * b2;
        gate[b * DIM + o2] = 1.f / (1.f + expf(-att));
    }
}

// ---------------------------------------------------------------------------
// Kernel 4: fused channel gate + per-token spatial gate + output.
//   xg = x * gate[b,:];  s = dot(xg, Ws[b,:]) + bs[b];  out = xg * sigmoid(s)
// One wave (32 lanes) per token, 8 tokens per block. gate and gate*Ws staged
// in LDS once per block. Wave-level shuffle reduction for the 768-dot.
// ---------------------------------------------------------------------------
__global__ __launch_bounds__(256) void apply_gate_kernel(
    const float* __restrict__ x,     // [B, N, D]
    const float* __restrict__ gate,  // [B, D]
    const float* __restrict__ Ws,    // [B, D]
    const float* __restrict__ bs,    // [B]
    float* __restrict__ out)         // [B, N, D]
{
    __shared__ float sg[DIM];   // gate
    __shared__ float sw[DIM];   // gate * Ws

    const int t = threadIdx.x;
    const int chunks = NTOK / 8;                 // 128
    const int b     = blockIdx.x / chunks;
    const int chunk = blockIdx.x - b * chunks;

    for (int j = t; j < DIM; j += 256) {
        const float g = gate[b * DIM + j];
        sg[j] = g;
        sw[j] = g * Ws[b * DIM + j];
    }
    __syncthreads();

    const int wave = t >> 5;
    const int lane = t & 31;
    const int nn   = chunk * 8 + wave;
    const float* xr   = x   + ((size_t)b * NTOK + nn) * DIM;
    float*       outr = out + ((size_t)b * NTOK + nn) * DIM;

    float xg[DIM / 32];
    float partial = 0.f;
    #pragma unroll
    for (int j = 0; j < DIM / 32; ++j) {
        const int d  = lane + 32 * j;
        const float xv = xr[d];
        xg[j] = xv * sg[d];
        partial += xv * sw[d];       // = xg * Ws since sw = gate*Ws
    }
    // wave32 butterfly reduction
    #pragma unroll
    for (int off = 16; off > 0; off >>= 1)
        partial += __shfl_xor(partial, off, 32);

    const float s   = partial + bs[b];
    const float sig = 1.f / (1.f + expf(-s));
    #pragma unroll
    for (int j = 0; j < DIM / 32; ++j)
        outr[lane + 32 * j] = xg[j] * sig;
}

// ---------------------------------------------------------------------------
// Launch
// ---------------------------------------------------------------------------
extern "C" void kernel_launch(void* const* d_in, const int* in_sizes, int n_in,
                              void* d_out, int out_size, void* d_ws, size_t ws_size,
                              hipStream_t stream) {
    const float* x   = (const float*)d_in[0];
    const float* mu  = (const float*)d_in[1];
    const float* Wg1 = (const float*)d_in[2];
    const float* bg1 = (const float*)d_in[3];
    const float* Bg1 = (const float*)d_in[4];
    const float* bb1 = (const float*)d_in[5];
    const float* Wg2 = (const float*)d_in[6];
    const float* bg2 = (const float*)d_in[7];
    const float* Bg2 = (const float*)d_in[8];
    const float* bb2 = (const float*)d_in[9];
    const float* Wgs = (const float*)d_in[10];
    const float* bgs = (const float*)d_in[11];
    const float* Bgs = (const float*)d_in[12];
    const float* bbs = (const float*)d_in[13];
    float* out = (float*)d_out;

    // Workspace layout (floats); total ~4.9 MB
    float* ws     = (float*)d_ws;
    float* W1full = ws;                           // 16*36864
    float* W2full = W1full + BSZ * NT_BIG;        // 16*36864
    float* Wsfull = W2full + BSZ * NT_BIG;        // 16*768
    float* avg    = Wsfull + BSZ * DIM;           // 16*768
    float* mxp    = avg    + BSZ * DIM;           // 16*768
    float* gatep  = mxp    + BSZ * DIM;           // 16*768
    float* bsv    = gatep  + BSZ * DIM;           // 16

    // Hyper-network weight generation (WMMA f32, memory-bound: 226 MB stream)
    hyper_gemm_wmma<<<NT_BIG / 128, 256, 0, stream>>>(mu, Wg1, bg1, W1full, NT_BIG);
    hyper_gemm_wmma<<<NT_BIG / 128, 256, 0, stream>>>(mu, Wg2, bg2, W2full, NT_BIG);
    hyper_gemm_wmma<<<DIM    / 128, 256, 0, stream>>>(mu, Wgs, bgs, Wsfull, DIM);

    // Pooling over tokens (reads x; stays L2-resident for the final kernel)
    pool_kernel<<<(BSZ * DIM) / 256, 256, 0, stream>>>(x, avg, mxp);

    // Per-sample dynamic MLP -> channel gate + spatial bias
    mlp_gate_kernel<<<BSZ, 256, 0, stream>>>(mu, W1full, W2full,
                                             Bg1, bb1, Bg2, bb2, Bgs, bbs,
                                             avg, mxp, gatep, bsv);

    // Fused channel-gate * spatial-gate output pass
    apply_gate_kernel<<<BSZ * (NTOK / 8), 256, 0, stream>>>(x, gatep, Wsfull, bsv, out);
}